// Stitcher_16527034155151
// MI455X (gfx1250) — compile-verified
//
#include <hip/hip_runtime.h>
#include <cstdint>

// ---------------------------------------------------------------------------
// Sparse COO union with max-|v| dedup into dense 8192x8192 fp32.
//
// key encoding: key = (bits[30:0] << 1) | sign_bit
//   - monotonic in |v| for finite floats (abs bits as uint are monotonic)
//   - losslessly invertible: bits = (key >> 1) | (key << 31)
//   - key == 0  <=>  +0.0  (matches "untouched position -> 0")
//
// Pipeline: init keys to 0  ->  atomic_max_u32 scatter  ->  in-place decode.
// Memory-bound (~1 GB HBM traffic @ 23.3 TB/s ~ 45 us). WMMA not applicable.
// ---------------------------------------------------------------------------

typedef __attribute__((ext_vector_type(4))) unsigned int v4u;

#define C_LOG2 13   // C = 8192

// ---- kernel 1: zero the key array (128-bit regular-temporal stores so the
// freshly written lines are L2-resident when the atomics hit them) ----------
__global__ void stitch_init_keys(unsigned int* __restrict__ buf, long long n) {
    const long long n4 = n >> 2;
    const long long i  = (long long)blockIdx.x * blockDim.x + threadIdx.x;
    v4u* p = (v4u*)buf;
    if (i < n4) {
        v4u z = {0u, 0u, 0u, 0u};
        p[i] = z;                      // RT store: keep line in L2 for atomics
    }
    if (i == 0) {                      // scalar tail (n % 4), normally empty
        for (long long t = (n4 << 2); t < n; ++t) buf[t] = 0u;
    }
}

// ---- kernel 2: scatter with 32-bit no-return atomic max -------------------
// Streaming reads use non-temporal hints so the 160 MB of index/value data
// does not evict the 256 MB atomic target range from the 192 MB L2.
__global__ void stitch_scatter_max(const float*     __restrict__ vals,
                                   const long long* __restrict__ rows,
                                   const long long* __restrict__ cols,
                                   long long n,
                                   unsigned int*    __restrict__ keys) {
    const long long stride = (long long)gridDim.x * blockDim.x;  // per slot
    const long long group  = stride * 4;

    for (long long base = (long long)blockIdx.x * blockDim.x + threadIdx.x;
         base < n; base += group) {
        // software prefetch of the next group of this thread's stream
        const long long pf = base + group;
        if (pf < n) {
            __builtin_prefetch(rows + pf, 0, 1);   // global_prefetch_b8
            __builtin_prefetch(cols + pf, 0, 1);
            __builtin_prefetch(vals + pf, 0, 1);
        }

        long long r[4], c[4];
        float     v[4];
        bool      ok[4];
#pragma unroll
        for (int k = 0; k < 4; ++k) {              // issue all loads first (MLP)
            const long long i = base + (long long)k * stride;
            ok[k] = (i < n);
            if (ok[k]) {
                r[k] = __builtin_nontemporal_load(rows + i);  // th:TH_LOAD_NT
                c[k] = __builtin_nontemporal_load(cols + i);
                v[k] = __builtin_nontemporal_load(vals + i);
            }
        }
#pragma unroll
        for (int k = 0; k < 4; ++k) {
            if (ok[k]) {
                const unsigned int b    = __float_as_uint(v[k]);
                const unsigned int key  = ((b & 0x7fffffffu) << 1) | (b >> 31);
                const unsigned int flat = ((unsigned int)r[k] << C_LOG2)
                                        + (unsigned int)c[k];
                atomicMax(keys + flat, key);       // global_atomic_max_u32 (no ret)
            }
        }
    }
}

// ---- kernel 3: in-place decode key -> float (pure bit rotate) -------------
__global__ void stitch_decode(unsigned int* __restrict__ buf, long long n) {
    const long long n4 = n >> 2;
    const long long i  = (long long)blockIdx.x * blockDim.x + threadIdx.x;
    v4u* p = (v4u*)buf;
    if (i < n4) {
        v4u k = __builtin_nontemporal_load(p + i);  // last use of key data
        v4u o;
#pragma unroll
        for (int j = 0; j < 4; ++j) o[j] = (k[j] >> 1) | (k[j] << 31);
        __builtin_nontemporal_store(o, p + i);      // final output, stream out
    }
    if (i == 0) {
        for (long long t = (n4 << 2); t < n; ++t) {
            const unsigned int k = buf[t];
            buf[t] = (k >> 1) | (k << 31);
        }
    }
}

extern "C" void kernel_launch(void* const* d_in, const int* in_sizes, int n_in,
                              void* d_out, int out_size, void* d_ws, size_t ws_size,
                              hipStream_t stream) {
    (void)n_in; (void)d_ws; (void)ws_size;

    const float*     vals_e = (const float*)d_in[0];
    const float*     vals_n = (const float*)d_in[1];
    const long long* idx_e  = (const long long*)d_in[2];  // [2, NE] flat
    const long long* idx_n  = (const long long*)d_in[3];  // [2, NN] flat

    const long long ne = in_sizes[0];
    const long long nn = in_sizes[1];
    const long long n  = (long long)out_size;             // R*C = 64M

    unsigned int* keys = (unsigned int*)d_out;            // reuse output as key buf

    const int TB = 256;

    // 1) init 64M keys to zero (16M x 128-bit stores)
    const long long n4      = n >> 2;
    const unsigned  iblocks = (unsigned)((n4 + TB - 1) / TB);
    stitch_init_keys<<<iblocks, TB, 0, stream>>>(keys, n);

    // 2) scatter both streams (grid-stride, 4 entries/thread/iteration)
    const unsigned sblocks = 1024;
    stitch_scatter_max<<<sblocks, TB, 0, stream>>>(vals_e, idx_e, idx_e + ne, ne, keys);
    stitch_scatter_max<<<sblocks, TB, 0, stream>>>(vals_n, idx_n, idx_n + nn, nn, keys);

    // 3) decode keys -> floats in place
    stitch_decode<<<iblocks, TB, 0, stream>>>(keys, n);
}